// ModelNew_10522669875235
// MI455X (gfx1250) — compile-verified
//
#include <hip/hip_runtime.h>

// CSR neighbor max-aggregation for MI455X (gfx1250, wave32).
//
// Mapping: one wave per destination row. 32 lanes x float2 = one 256B
// feature row per coalesced global_load_b64.
//
// Index stream: col_idx[i+k] with i in an SGPR is a uniform load from
// read-only memory -> scalar cache (s_load_bN, KMcnt), keeping the vector
// memory pipe free for the 205MB feature gather.
//
// Pipeline: 8-edge groups; next group's column indices are fetched and its
// feature rows prefetched (global_prefetch_b8) while the current group's
// 8 independent b64 gathers (2KB/wave in flight) are consumed.

typedef float v2f __attribute__((ext_vector_type(2)));

#define FEAT_DIM 64
#define WPB 8                 // waves per block
#define TPB (WPB * 32)        // 256 threads = 8 wave32
#define G   8                 // edges per pipelined group

__global__ __launch_bounds__(TPB)
void csr_neighbor_max_kernel(const int* __restrict__ row_ptr,
                             const int* __restrict__ col_idx,
                             const float* __restrict__ feat,
                             float* __restrict__ out,
                             int n_nodes) {
    const int lane = threadIdx.x & 31;
    const int row  = blockIdx.x * WPB + (threadIdx.x >> 5);
    if (row >= n_nodes) return;   // row is wave-uniform: whole wave exits together

    // Row extents are wave-uniform: force into SGPRs so the edge loop runs
    // on the SALU and all derived addresses get scalar bases.
    int s = __builtin_amdgcn_readfirstlane(row_ptr[row]);
    int e = __builtin_amdgcn_readfirstlane(row_ptr[row + 1]);

    const v2f* __restrict__ feat2 = (const v2f*)feat;   // 32 x float2 per node row

    v2f acc;
    acc.x = -__builtin_inff();
    acc.y = -__builtin_inff();

    int i = s;
    if (i + G <= e) {
        // Prologue: fetch first group's columns (uniform -> scalar cache).
        int cc[G];
#pragma unroll
        for (int k = 0; k < G; ++k) cc[k] = col_idx[i + k];
        i += G;

        // Steady state: fetch next group's columns + prefetch their feature
        // rows while gathering/reducing the current group.
        for (; i + G <= e; i += G) {
            int nn[G];
#pragma unroll
            for (int k = 0; k < G; ++k) nn[k] = col_idx[i + k];
#pragma unroll
            for (int k = 0; k < G; ++k)
                __builtin_prefetch(&feat2[(size_t)nn[k] * 32], 0, 3);

            v2f v[G];
#pragma unroll
            for (int k = 0; k < G; ++k) v[k] = feat2[(size_t)cc[k] * 32 + lane];
#pragma unroll
            for (int k = 0; k < G; ++k) {
                acc.x = fmaxf(acc.x, v[k].x);
                acc.y = fmaxf(acc.y, v[k].y);
            }
#pragma unroll
            for (int k = 0; k < G; ++k) cc[k] = nn[k];
        }

        // Epilogue: drain the last full group.
        v2f v[G];
#pragma unroll
        for (int k = 0; k < G; ++k) v[k] = feat2[(size_t)cc[k] * 32 + lane];
#pragma unroll
        for (int k = 0; k < G; ++k) {
            acc.x = fmaxf(acc.x, v[k].x);
            acc.y = fmaxf(acc.y, v[k].y);
        }
    }

    // Scalar tail (< G edges remain); column loads stay uniform.
    for (; i < e; ++i) {
        const int c = col_idx[i];
        v2f v = feat2[(size_t)c * 32 + lane];
        acc.x = fmaxf(acc.x, v.x);
        acc.y = fmaxf(acc.y, v.y);
    }

    // Non-temporal store: don't evict the L2-resident feature table with
    // the 12.8MB streaming output.
    v2f* op = (v2f*)out + (size_t)row * 32 + lane;
    __builtin_nontemporal_store(acc, op);
}

extern "C" void kernel_launch(void* const* d_in, const int* in_sizes, int n_in,
                              void* d_out, int out_size, void* d_ws, size_t ws_size,
                              hipStream_t stream) {
    const int*   row_ptr = (const int*)d_in[0];    // [N+1]
    const int*   col_idx = (const int*)d_in[1];    // [E]
    const float* feat    = (const float*)d_in[2];  // [N, 64]
    float*       out     = (float*)d_out;          // [N, 64]

    const int n_nodes = in_sizes[0] - 1;
    const int blocks  = (n_nodes + WPB - 1) / WPB;

    hipLaunchKernelGGL(csr_neighbor_max_kernel, dim3(blocks), dim3(TPB), 0, stream,
                       row_ptr, col_idx, feat, out, n_nodes);
}